// LabelSmoothingCE_28381143892403
// MI455X (gfx1250) — compile-verified
//
#include <hip/hip_runtime.h>
#include <stdint.h>

#define NUM_CLASSES 9
#define TILE_ROWS   256
#define TILE_ELEMS  (TILE_ROWS * NUM_CLASSES)   // 2304 floats = 9216 B
#define TILE_V4     (TILE_ELEMS / 4)            // 576 float4
#define LOADERS     192                         // threads issuing async loads
#define V4_PER_THR  (TILE_V4 / LOADERS)         // exactly 3 per loader thread
#define BLOCK       256
#define MAX_BLOCKS  2048

// ---- CDNA5 async global->LDS path (guarded; fallback = vector loads) ----
#if defined(__HIP_DEVICE_COMPILE__) && \
    __has_builtin(__builtin_amdgcn_global_load_async_to_lds_b128) && \
    __has_builtin(__builtin_amdgcn_s_wait_asynccnt)
#define USE_ASYNC 1
typedef int v4i __attribute__((vector_size(16)));
typedef __attribute__((address_space(1))) v4i* gv4i_p;   // global int4*
typedef __attribute__((address_space(3))) v4i* lv4i_p;   // LDS int4*

__device__ __forceinline__ gv4i_p as_g4(const void* p) {
  return (gv4i_p)(uint64_t)p;                 // flat addr == global VA
}
__device__ __forceinline__ lv4i_p as_l4(void* p) {
  // low 32 bits of a flat LDS-aperture address are the LDS byte offset
  return (lv4i_p)(uint32_t)(uint64_t)p;
}
#endif

// Issue one full tile (2304 floats) of async global->LDS traffic.
// Exactly LOADERS threads issue exactly V4_PER_THR b128 ops each, so every
// issuing wave's ASYNCcnt delta is uniform (required for per-wave waits).
__device__ __forceinline__ void issue_tile(const float* __restrict__ gsrc,
                                           float* __restrict__ ldst, int tid) {
  if (tid < LOADERS) {
#if USE_ASYNC
    #pragma unroll
    for (int j = 0; j < V4_PER_THR; ++j) {
      const int i4 = tid + j * LOADERS;
      __builtin_amdgcn_global_load_async_to_lds_b128(
          as_g4((const float4*)gsrc + i4),
          as_l4((float4*)ldst + i4), 0, 0);
    }
#else
    #pragma unroll
    for (int j = 0; j < V4_PER_THR; ++j) {
      const int i4 = tid + j * LOADERS;
      ((float4*)ldst)[i4] = ((const float4*)gsrc)[i4];
    }
#endif
  }
}

__global__ __launch_bounds__(BLOCK)
void ce_main_kernel(const float* __restrict__ logits,
                    const int*   __restrict__ targets,
                    float*       __restrict__ partials,
                    int batch, int numTiles) {
  __shared__ __align__(16) float sh[2][TILE_ELEMS];   // double buffer, 18.4 KB
  __shared__ float wred[BLOCK / 32];

  const int tid = threadIdx.x;
  const int fullTiles = batch / TILE_ROWS;    // tiles with all 256 rows valid
  float acc = 0.0f;

  const float base = 0.1f / 9.0f;             // SMOOTHING / C
  const float c1   = 0.8f - base;             // main_w - base
  const float wA   = 1.0f - base;             // constant part of Sum(w)

  // ---- software pipeline prologue: stage first tile into buf 0 ----
  const int t0 = blockIdx.x;
  int cur = 0;
  if (t0 < numTiles && t0 < fullTiles)
    issue_tile(logits + (size_t)t0 * TILE_ELEMS, sh[0], tid);

  for (int tile = t0; tile < numTiles; tile += gridDim.x) {
    // Issue NEXT tile into the other buffer (its last readers were in the
    // previous iteration; a barrier has intervened).
    const int  nxt      = tile + gridDim.x;
    const bool nextFull = (nxt < numTiles) && (nxt < fullTiles);
    if (nextFull)
      issue_tile(logits + (size_t)nxt * TILE_ELEMS, sh[cur ^ 1], tid);

    const long long tbase = (long long)tile * TILE_ROWS;
    const int rowsHere = min(TILE_ROWS, batch - (int)tbase);
    if (rowsHere < TILE_ROWS) {
      // Globally-last (partial) tile: never async-issued; fill synchronously.
      const float* gsrc = logits + tbase * NUM_CLASSES;
      const int elems = rowsHere * NUM_CLASSES;
      for (int idx = tid; idx < elems; idx += BLOCK) sh[cur][idx] = gsrc[idx];
    }

#if USE_ASYNC
    // Async loads complete FIFO per wave: leaving V4_PER_THR outstanding
    // means the CURRENT tile's loads are done, the next tile's still fly.
    if (nextFull) __builtin_amdgcn_s_wait_asynccnt(V4_PER_THR);
    else          __builtin_amdgcn_s_wait_asynccnt(0);
#endif
    __syncthreads();                          // cross-wave visibility of sh[cur]

    const int row = (int)tbase + tid;
    if (row < batch) {
      const int t = targets[row];
      float x[NUM_CLASSES];
      #pragma unroll
      for (int k = 0; k < NUM_CLASSES; ++k) x[k] = sh[cur][tid * NUM_CLASSES + k];

      // logsumexp
      float m = x[0];
      #pragma unroll
      for (int k = 1; k < NUM_CLASSES; ++k) m = fmaxf(m, x[k]);
      float se = 0.0f;
      #pragma unroll
      for (int k = 0; k < NUM_CLASSES; ++k) se += __expf(x[k] - m);
      const float lse = m + __logf(se);

      // w.x with w_k = base + [k==t]*(0.8-base) + [k>t]*0.1*2^(t-k)
      float sumx = 0.0f, xt = 0.0f, geo = 0.0f;
      #pragma unroll
      for (int k = 0; k < NUM_CLASSES; ++k) {
        sumx += x[k];
        xt = (k == t) ? x[k] : xt;
        geo += (k > t) ? ldexpf(x[k], t - k) : 0.0f;   // v_ldexp_f32, exact
      }
      const float wsum = wA - ldexpf(0.1f, t - 8);     // analytic Sum(w)
      const float dot  = fmaf(base, sumx, fmaf(c1, xt, 0.1f * geo));
      acc += lse - dot / wsum;
    }
    __syncthreads();   // all reads of sh[cur] done before it is re-issued
    cur ^= 1;
  }

  // wave32 reduction then cross-wave via LDS (deterministic order)
  #pragma unroll
  for (int off = 16; off > 0; off >>= 1) acc += __shfl_xor(acc, off, 32);
  if ((tid & 31) == 0) wred[tid >> 5] = acc;
  __syncthreads();
  if (tid == 0) {
    float b = 0.0f;
    #pragma unroll
    for (int i = 0; i < BLOCK / 32; ++i) b += wred[i];
    partials[blockIdx.x] = b;
  }
}

__global__ __launch_bounds__(256)
void ce_reduce_kernel(const float* __restrict__ partials, int n,
                      float* __restrict__ out, double scale) {
  __shared__ double s[256];
  const int tid = threadIdx.x;
  double a = 0.0;
  for (int i = tid; i < n; i += 256) a += (double)partials[i];
  s[tid] = a;
  __syncthreads();
  #pragma unroll
  for (int off = 128; off > 0; off >>= 1) {
    if (tid < off) s[tid] += s[tid + off];
    __syncthreads();
  }
  if (tid == 0) out[0] = (float)(s[0] * scale);
}

extern "C" void kernel_launch(void* const* d_in, const int* in_sizes, int n_in,
                              void* d_out, int out_size, void* d_ws, size_t ws_size,
                              hipStream_t stream) {
  const float* logits  = (const float*)d_in[0];
  const int*   targets = (const int*)d_in[1];
  const int batch = in_sizes[1];                       // targets count == BATCH
  const int numTiles = (batch + TILE_ROWS - 1) / TILE_ROWS;
  int nb = numTiles < MAX_BLOCKS ? numTiles : MAX_BLOCKS;
  float* partials = (float*)d_ws;

  ce_main_kernel<<<nb, BLOCK, 0, stream>>>(logits, targets, partials, batch, numTiles);
  ce_reduce_kernel<<<1, 256, 0, stream>>>(partials, nb, (float*)d_out,
                                          1.0 / (double)batch);
}